// ACDC2D_35038343201374
// MI455X (gfx1250) — compile-verified
//
#include <hip/hip_runtime.h>

typedef __attribute__((ext_vector_type(2))) float v2f;
typedef __attribute__((ext_vector_type(4))) float v4f;
typedef __attribute__((ext_vector_type(8))) float v8f;

#define HD 192
#define WD 192
#define HW (HD * WD)       // 36864
#define CD 64
#define BD 16

// d_ws layout (float offsets)
#define WS_M   0           // 64 x v2f  = 128 floats  (m = ifft(D))
#define WS_B2  128         // 64 floats (bias[perm[o]]/8)
#define WS_W   256         // 144*8*32 v2f = 294912 bytes (B-matrix swizzled weights)

#define NKK   144          // K-steps: K = 576 = 144 * 4
#define LDS_BYTES (NKK * 8 * 32 * 8)   // 294912 B of LDS weights

#define WMMA_F32(a, b, c) __builtin_amdgcn_wmma_f32_16x16x4_f32( \
    false, (a), false, (b), (short)0, (c), false, false)

// ---------------- prep 1: m[n] = ifft(D)[n], b2[o] = bias[perm[o]]/8 ----------------
__global__ void acdc_prep1(const float* __restrict__ D, const float* __restrict__ bias,
                           const int* __restrict__ perm, float* __restrict__ ws) {
  int n = threadIdx.x;                       // 0..63
  float re = 0.f, im = 0.f;
  for (int j = 0; j < 64; ++j) {
    float th = 0.098174770424681f * (float)((j * n) & 63);   // 2*pi/64 * (j*n mod 64)
    float s, c;
    __sincosf(th, &s, &c);
    float d = D[j];
    re += d * c;
    im += d * s;
  }
  v2f mv; mv.x = re * (1.f / 64.f); mv.y = im * (1.f / 64.f);
  ((v2f*)(ws + WS_M))[n] = mv;
  ws[WS_B2 + n] = bias[perm[n]] * 0.125f;
}

// ---------------- prep 2: fold conv_w, A, perm, m into swizzled B-matrix weights ----
// slot = kk*256 + n*32 + lane : v2f { W[kk*4 + (lane<16?0:1)][N] , W[kk*4 + (lane<16?2:3)][N] }
// N = n*16 + (lane&15);  N<64 -> Re(Wc[N][k]) ; N>=64 -> Im(Wc[N-64][k])
// k = t*64 + cin  (t = 3x3 tap, cin = input channel)
__global__ void acdc_prep2(const float* __restrict__ conv_w, const float* __restrict__ A,
                           const int* __restrict__ perm, float* __restrict__ ws) {
  int id   = blockIdx.x * blockDim.x + threadIdx.x;   // 0..36863
  int lane = id & 31;
  int n    = (id >> 5) & 7;
  int kk   = id >> 8;
  int N    = n * 16 + (lane & 15);
  int o    = N & 63;
  int po   = perm[o];
  bool is_im = (N >= 64);
  const v2f* m = (const v2f*)(ws + WS_M);

  int ks[2];
  ks[0] = kk * 4 + ((lane < 16) ? 0 : 1);
  ks[1] = kk * 4 + ((lane < 16) ? 2 : 3);
  float outv[2];
  for (int q = 0; q < 2; ++q) {
    int k = ks[q];
    int t = k >> 6;            // tap 0..8
    int cin = k & 63;
    int g = cin >> 3, ci = cin & 7;
    float ar = 0.f, ai = 0.f;
    for (int i = 0; i < 8; ++i) {
      int c = g * 8 + i;                       // conv out-channel in this group
      v2f mm = m[(po - c) & 63];
      float s = A[c] * 0.125f;
      float cw = conv_w[(c * 8 + ci) * 9 + t];
      ar += mm.x * s * cw;
      ai += mm.y * s * cw;
    }
    outv[q] = is_im ? ai : ar;
  }
  v2f r; r.x = outv[0]; r.y = outv[1];
  ((v2f*)(ws + WS_W))[id] = r;
}

// ---------------- main: implicit-GEMM conv via V_WMMA_F32_16X16X4_F32 ----------------
// supertile = 32 spatial (along w) x 128 N (4 real + 4 imag N-tiles), 16 WMMA / K-step
__device__ __forceinline__ void store_tile(const v8f accR[4], const v8f accI[4],
                                           float* __restrict__ out,
                                           int b, int h, int wbase, int m16) {
#pragma unroll
  for (int n = 0; n < 4; ++n) {
    int o = n * 16 + m16;
    v4f* op = (v4f*)((v2f*)out + ((b * CD + o) * HD + h) * WD + wbase);
#pragma unroll
    for (int v = 0; v < 8; v += 2) {
      v4f val;
      val.x = accR[n][v];     val.y = accI[n][v];
      val.z = accR[n][v + 1]; val.w = accI[n][v + 1];
      __builtin_nontemporal_store(val, op + (v >> 1));
    }
  }
}

__global__ __launch_bounds__(512)
void acdc_main(const float* __restrict__ x, const float* __restrict__ ws,
               float* __restrict__ out) {
  extern __shared__ char smem_raw[];
  v2f* wlds = (v2f*)smem_raw;

  // cooperative weight load: 294912 B = 18432 x b128
  {
    const v4f* src = (const v4f*)(ws + WS_W);
    v4f* dst = (v4f*)smem_raw;
    for (int i = threadIdx.x; i < LDS_BYTES / 16; i += 512) dst[i] = src[i];
  }
  __syncthreads();

  const int lane = threadIdx.x & 31;
  const int wv   = threadIdx.x >> 5;   // wave 0..15
  const int m16  = lane & 15;          // spatial pos within tile / N within tile
  const int hs   = lane >> 4;          // half-wave select

  float b2v[4];
#pragma unroll
  for (int n = 0; n < 4; ++n) b2v[n] = ws[WS_B2 + n * 16 + m16];

  for (int it = 0; it < 2; ++it) {
    int st = (blockIdx.x * 16 + wv) * 2 + it;     // 576 blocks * 16 waves * 2 = 18432
    int w0 = (st % 6) * 32;
    int h  = (st / 6) % HD;
    int b  = st / (6 * HD);

    v8f accR0[4], accI0[4], accR1[4], accI1[4];
#pragma unroll
    for (int n = 0; n < 4; ++n) {
      v8f r;
#pragma unroll
      for (int v = 0; v < 8; ++v) r[v] = b2v[n];    // bias -> real part
      accR0[n] = r;
      accR1[n] = r;
      v8f z = {};
      accI0[n] = z;
      accI1[n] = z;
    }

#pragma unroll
    for (int t = 0; t < 9; ++t) {
      const int dh = t / 3 - 1, dw = t % 3 - 1;
      int hh = h + dh;
      int hc = min(max(hh, 0), HD - 1);
      int rok = (hh == hc);
      int wq0 = w0 + m16 + dw;
      int wq1 = wq0 + 16;
      int wc0 = min(max(wq0, 0), WD - 1);
      int wc1 = min(max(wq1, 0), WD - 1);
      float ok0 = (rok & (wq0 == wc0)) ? 1.f : 0.f;   // zero-pad masks, branchless
      float ok1 = (rok & (wq1 == wc1)) ? 1.f : 0.f;
      // A-matrix channels for this lane: base + {0,1} (lanes 0-15) or base+2 + {0,1}
      const float* xrow = x + ((b * CD + hs * 2) * HD + hc) * WD;
      const float* xp0 = xrow + wc0;
      const float* xp1 = xrow + wc1;
#pragma unroll 4
      for (int cs = 0; cs < 16; ++cs) {
        v2f a0, a1;
        a0.x = xp0[0]  * ok0;
        a0.y = xp0[HW] * ok0;
        a1.x = xp1[0]  * ok1;
        a1.y = xp1[HW] * ok1;
        const v2f* wp = &wlds[(t * 16 + cs) * 256 + lane];
#pragma unroll
        for (int n = 0; n < 4; ++n) {
          v2f bvR = wp[n * 32];
          v2f bvI = wp[(n + 4) * 32];
          accR0[n] = WMMA_F32(a0, bvR, accR0[n]);
          accI0[n] = WMMA_F32(a0, bvI, accI0[n]);
          accR1[n] = WMMA_F32(a1, bvR, accR1[n]);
          accI1[n] = WMMA_F32(a1, bvI, accI1[n]);
        }
        xp0 += 4 * HW;
        xp1 += 4 * HW;
      }
    }

    store_tile(accR0, accI0, out, b, h, w0 + hs * 8, m16);
    store_tile(accR1, accI1, out, b, h, w0 + 16 + hs * 8, m16);
  }
}

extern "C" void kernel_launch(void* const* d_in, const int* in_sizes, int n_in,
                              void* d_out, int out_size, void* d_ws, size_t ws_size,
                              hipStream_t stream) {
  (void)in_sizes; (void)n_in; (void)out_size; (void)ws_size;
  const float* x      = (const float*)d_in[0];
  const float* conv_w = (const float*)d_in[1];
  const float* A      = (const float*)d_in[2];
  const float* D      = (const float*)d_in[3];
  const float* bias   = (const float*)d_in[4];
  const int*   perm   = (const int*)d_in[5];
  float* ws  = (float*)d_ws;
  float* out = (float*)d_out;

  acdc_prep1<<<1, 64, 0, stream>>>(D, bias, perm, ws);
  acdc_prep2<<<144, 256, 0, stream>>>(conv_w, A, perm, ws);
  acdc_main<<<576, 512, LDS_BYTES, stream>>>(x, ws, out);
}